// FTNet_11751030522504
// MI455X (gfx1250) — compile-verified
//
#include <hip/hip_runtime.h>

typedef __attribute__((ext_vector_type(2))) float v2f;
typedef __attribute__((ext_vector_type(8))) float v8f;

#define N_CORES 32
#define DIM 16
#define LATENT 100
#define MAXOD 64
#define KC1 4            // DIM / 4 K-chunks for layer 1
#define KC2 25           // LATENT / 4 K-chunks for layers 2/3 (exact, no pad)
#define NT 7             // ceil(LATENT/16) N tiles (last partial, zero padded)
#define H_STRIDE 114     // LDS row stride in floats: even (b64 align) + conflict-free
#define WPB 2            // waves per block
#define MT 2             // 16-row batch tiles per wave (B-fragment reuse)

__global__ __launch_bounds__(WPB * 32)
void ftnet_tt_kernel(const float* __restrict__ x,
                     const float* __restrict__ W1, const float* __restrict__ b1,
                     const float* __restrict__ W2, const float* __restrict__ b2,
                     const float* __restrict__ W3, const float* __restrict__ b3,
                     float* __restrict__ out, int batch)
{
    // [wave][tile][buffer 0/1][16 rows x H_STRIDE]
    __shared__ float lds[WPB][MT][2][16 * H_STRIDE];

    const int tid  = threadIdx.x;
    const int wave = tid >> 5;
    const int lane = tid & 31;
    const int m    = lane & 15;   // A-layout row / B&C-layout column-in-tile
    const int half = lane >> 4;   // selects K pair within a K=4 chunk

    const int tileBase = (blockIdx.x * WPB + wave) * (MT * 16);
    if (tileBase + MT * 16 > batch) return;   // wave-uniform guard (EXEC stays full)

    const v8f vzero = {};
    float ret[MT][8];             // tensor-train chain state (valid in lanes 0..15)
    #pragma unroll
    for (int t = 0; t < MT; ++t)
        #pragma unroll
        for (int k = 0; k < 8; ++k) ret[t][k] = 1.0f;

    for (int core = 0; core < N_CORES; ++core) {
        const float* W1c = W1 + (size_t)core * DIM * LATENT;
        const float* b1c = b1 + (size_t)core * LATENT;
        const float* W2c = W2 + (size_t)core * LATENT * LATENT;
        const float* b2c = b2 + (size_t)core * LATENT;
        const float* W3c = W3 + (size_t)core * LATENT * MAXOD;
        const float* b3c = b3 + (size_t)core * MAXOD;

        if (core + 1 < N_CORES)   // warm L2/near caches for the next core's big weight
            __builtin_prefetch(W2 + (size_t)(core + 1) * LATENT * LATENT, 0, 1);

        // ---------------- Layer 1: (16x16) @ (16x100) ----------------
        v8f acc[MT][NT];
        #pragma unroll
        for (int t = 0; t < MT; ++t)
            #pragma unroll
            for (int nt = 0; nt < NT; ++nt) acc[t][nt] = vzero;

        #pragma unroll
        for (int kc = 0; kc < KC1; ++kc) {
            const int kb = kc * 4 + half * 2;
            v2f a[MT];
            #pragma unroll
            for (int t = 0; t < MT; ++t) {
                const float* xp = x + (size_t)(tileBase + t * 16 + m) * (N_CORES * DIM)
                                    + core * DIM + kb;
                a[t] = *(const v2f*)xp;
            }
            #pragma unroll
            for (int nt = 0; nt < NT; ++nt) {
                const int n = nt * 16 + m;
                v2f bfr;
                if (n < LATENT) {
                    const float* wp = W1c + (size_t)kb * LATENT + n;
                    bfr.x = wp[0];
                    bfr.y = wp[LATENT];
                } else {
                    bfr.x = 0.0f; bfr.y = 0.0f;
                }
                #pragma unroll
                for (int t = 0; t < MT; ++t)
                    acc[t][nt] = __builtin_amdgcn_wmma_f32_16x16x4_f32(
                        false, a[t], false, bfr, (short)0, acc[t][nt], false, false);
            }
        }
        // bias + relu, scatter C/D layout -> row-major h in LDS buffer 0
        #pragma unroll
        for (int nt = 0; nt < NT; ++nt) {
            const int n = nt * 16 + m;
            if (n < LATENT) {
                const float bias = b1c[n];
                #pragma unroll
                for (int t = 0; t < MT; ++t) {
                    float* h = lds[wave][t][0];
                    #pragma unroll
                    for (int r = 0; r < 8; ++r) {
                        float v = acc[t][nt][r] + bias;
                        h[(r + 8 * half) * H_STRIDE + n] = v > 0.0f ? v : 0.0f;
                    }
                }
            }
        }

        // ---------------- Layer 2: (16x100) @ (100x100) ----------------
        #pragma unroll
        for (int t = 0; t < MT; ++t)
            #pragma unroll
            for (int nt = 0; nt < NT; ++nt) acc[t][nt] = vzero;

        for (int kc = 0; kc < KC2; ++kc) {
            const int kb = kc * 4 + half * 2;
            v2f a[MT];
            #pragma unroll
            for (int t = 0; t < MT; ++t)
                a[t] = *(const v2f*)&lds[wave][t][0][m * H_STRIDE + kb];
            #pragma unroll
            for (int nt = 0; nt < NT; ++nt) {
                const int n = nt * 16 + m;
                v2f bfr;
                if (n < LATENT) {
                    const float* wp = W2c + (size_t)kb * LATENT + n;
                    bfr.x = wp[0];
                    bfr.y = wp[LATENT];
                } else {
                    bfr.x = 0.0f; bfr.y = 0.0f;
                }
                #pragma unroll
                for (int t = 0; t < MT; ++t)
                    acc[t][nt] = __builtin_amdgcn_wmma_f32_16x16x4_f32(
                        false, a[t], false, bfr, (short)0, acc[t][nt], false, false);
            }
        }
        #pragma unroll
        for (int nt = 0; nt < NT; ++nt) {
            const int n = nt * 16 + m;
            if (n < LATENT) {
                const float bias = b2c[n];
                #pragma unroll
                for (int t = 0; t < MT; ++t) {
                    float* h = lds[wave][t][1];
                    #pragma unroll
                    for (int r = 0; r < 8; ++r) {
                        float v = acc[t][nt][r] + bias;
                        h[(r + 8 * half) * H_STRIDE + n] = v > 0.0f ? v : 0.0f;
                    }
                }
            }
        }

        // ---------------- Layer 3: (16x100) @ (100 x od) ----------------
        const int od  = (core == 0 || core == N_CORES - 1) ? 8 : MAXOD;
        const int nt3 = (od + 15) / 16;
        v8f acc3[MT][4];
        #pragma unroll
        for (int t = 0; t < MT; ++t)
            for (int nt = 0; nt < 4; ++nt) acc3[t][nt] = vzero;

        for (int kc = 0; kc < KC2; ++kc) {
            const int kb = kc * 4 + half * 2;
            v2f a[MT];
            #pragma unroll
            for (int t = 0; t < MT; ++t)
                a[t] = *(const v2f*)&lds[wave][t][1][m * H_STRIDE + kb];
            for (int nt = 0; nt < nt3; ++nt) {
                const int n = nt * 16 + m;          // n < 64 always: in-bounds of W3
                const float* wp = W3c + (size_t)kb * MAXOD + n;
                v2f bfr;
                bfr.x = wp[0];
                bfr.y = wp[MAXOD];
                #pragma unroll
                for (int t = 0; t < MT; ++t)
                    acc3[t][nt] = __builtin_amdgcn_wmma_f32_16x16x4_f32(
                        false, a[t], false, bfr, (short)0, acc3[t][nt], false, false);
            }
        }
        // bias + store G (row-major, od columns) into buffer 0 (h1 dead now)
        for (int nt = 0; nt < nt3; ++nt) {
            const int n = nt * 16 + m;
            if (n < od) {
                const float bias = b3c[n];
                #pragma unroll
                for (int t = 0; t < MT; ++t) {
                    float* g = lds[wave][t][0];
                    #pragma unroll
                    for (int r = 0; r < 8; ++r)
                        g[(r + 8 * half) * H_STRIDE + n] = acc3[t][nt][r] + bias;
                }
            }
        }

        // ---------------- tensor-train chain (lanes 0..15, one batch row each) ----
        if (lane < 16) {
            #pragma unroll
            for (int t = 0; t < MT; ++t) {
                const float* g = &lds[wave][t][0][m * H_STRIDE];
                if (core == 0) {
                    // ret = softmax(row of 8)
                    float v[8];
                    float mx = g[0];
                    #pragma unroll
                    for (int k = 0; k < 8; ++k) { v[k] = g[k]; mx = fmaxf(mx, v[k]); }
                    float s = 0.0f;
                    #pragma unroll
                    for (int k = 0; k < 8; ++k) { v[k] = __expf(v[k] - mx); s += v[k]; }
                    const float inv = 1.0f / s;
                    #pragma unroll
                    for (int k = 0; k < 8; ++k) ret[t][k] = v[k] * inv;
                } else if (core == N_CORES - 1) {
                    // final (1x8) @ (8x1)
                    float f = 0.0f;
                    #pragma unroll
                    for (int k = 0; k < 8; ++k) f += ret[t][k] * g[k];
                    out[tileBase + t * 16 + m] = f;
                } else {
                    // (1x8) @ (8x8), G[k][n] = g[k*8+n]
                    float nv[8];
                    #pragma unroll
                    for (int n = 0; n < 8; ++n) {
                        float s = 0.0f;
                        #pragma unroll
                        for (int k = 0; k < 8; ++k) s += ret[t][k] * g[k * 8 + n];
                        nv[n] = s;
                    }
                    #pragma unroll
                    for (int k = 0; k < 8; ++k) ret[t][k] = nv[k];
                }
            }
        }
    }
}

extern "C" void kernel_launch(void* const* d_in, const int* in_sizes, int n_in,
                              void* d_out, int out_size, void* d_ws, size_t ws_size,
                              hipStream_t stream) {
    const float* x  = (const float*)d_in[0];
    const float* W1 = (const float*)d_in[1];
    const float* b1 = (const float*)d_in[2];
    const float* W2 = (const float*)d_in[3];
    const float* b2 = (const float*)d_in[4];
    const float* W3 = (const float*)d_in[5];
    const float* b3 = (const float*)d_in[6];
    float* out = (float*)d_out;

    const int batch = in_sizes[0] / (N_CORES * DIM);
    const int rowsPerBlock = WPB * MT * 16;
    const int grid = (batch + rowsPerBlock - 1) / rowsPerBlock;

    ftnet_tt_kernel<<<grid, WPB * 32, 0, stream>>>(x, W1, b1, W2, b2, W3, b3, out, batch);
}